// QLLMMixer_55482387529791
// MI455X (gfx1250) — compile-verified
//
#include <hip/hip_runtime.h>
#include <math.h>

// QLLMMixer fused gather/softmax/mix kernel for MI455X (gfx1250, wave32).
//
// Strategy: selective-gather (only ~630B of each 5060B state row is needed),
// 16 lanes per row (lane == agent), butterfly reductions at width 16,
// async global->LDS prefetch of the q row overlapped with the gather phase.

#define N_AGENTS   11
#define OBS_DIM    115
#define STATE_DIM  (N_AGENTS * OBS_DIM)   // 1265
#define LANES_PER_ROW 16
#define THREADS    256
#define ROWS_PER_BLOCK (THREADS / LANES_PER_ROW)  // 16

// gfx1250 async global->LDS path (CDNA5 ISA).
#if defined(__gfx1250__) && \
    __has_builtin(__builtin_amdgcn_global_load_async_to_lds_b32) && \
    __has_builtin(__builtin_amdgcn_s_wait_asynccnt)
#define USE_ASYNC_LDS 1
#else
#define USE_ASYNC_LDS 0
#endif

// Builtin expects address-space-qualified int* operands (per hipcc diagnostic:
// first param is '__device__ int *', i.e. AS1 int*).
typedef __attribute__((address_space(1))) int gint_t;
typedef __attribute__((address_space(3))) int lint_t;

__global__ __launch_bounds__(THREADS)
void qmix_fused_kernel(const float* __restrict__ q,    // [B, 11]
                       const float* __restrict__ gs,   // [B, 1265]
                       float* __restrict__ out,        // [B]
                       int B)
{
    const int tid  = threadIdx.x;
    const int lane = tid & (LANES_PER_ROW - 1);
    const int grp  = tid >> 4;
    const int row  = blockIdx.x * ROWS_PER_BLOCK + grp;

#if USE_ASYNC_LDS
    __shared__ float q_s[ROWS_PER_BLOCK][N_AGENTS + 1];
    // Kick off the q-row fetch first; it lands in LDS while we do the
    // gather/argmax phase below. Same-wave producer/consumer: only
    // s_wait_asynccnt is needed, no workgroup barrier.
    if (row < B && lane < N_AGENTS) {
        __builtin_amdgcn_global_load_async_to_lds_b32(
            (gint_t*)(q + (size_t)row * N_AGENTS + lane),
            (lint_t*)&q_s[grp][lane],
            /*offset=*/0, /*cpol=*/0);
    }
#endif

    if (row >= B) return;   // uniform across the 16-lane group
    const float* g = gs + (size_t)row * STATE_DIM;

    // Row-uniform scalars: all 16 lanes hit the same cachelines (merged).
    const float bx     = g[88];
    const float by     = g[89];
    const float holder = g[95];
    const float gbx    = bx - 1.0f;
    const float ball_goal = sqrtf(gbx * gbx + by * by);
    const bool  valid  = (holder != 0.0f) && (ball_goal > 0.19f)
                                          && (ball_goal < 0.99f);

    // ---- per-agent gather: lane a handles agent a ----
    float dist  = __builtin_inff();
    float gdist = 0.0f;
    if (lane < N_AGENTS) {
        const float* obs = g + lane * OBS_DIM;
        // argmax over obs[97..107]; strict '>' gives first-max (jnp.argmax)
        float best = obs[97];
        int   bid  = 0;
#pragma unroll
        for (int j = 1; j < N_AGENTS; ++j) {
            float v = obs[97 + j];
            if (v > best) { best = v; bid = j; }
        }
        const float px = obs[2 * bid];
        const float py = obs[2 * bid + 1];
        const float dx = px - bx, dy = py - by;
        dist  = sqrtf(dx * dx + dy * dy);
        const float gx = px - 1.0f;
        gdist = sqrtf(gx * gx + py * py);
    }

    // ---- argmin(dist) over the row, lowest-index tie-break (jnp.argmin) ----
    float mval = dist;
    int   midx = lane;
#pragma unroll
    for (int off = 8; off >= 1; off >>= 1) {
        float ov = __shfl_xor(mval, off, LANES_PER_ROW);
        int   oi = __shfl_xor(midx, off, LANES_PER_ROW);
        if (ov < mval || (ov == mval && oi < midx)) { mval = ov; midx = oi; }
    }

    float combined = -__builtin_inff();
    if (lane < N_AGENTS)
        combined = valid ? ((lane == midx) ? 5.0f : 0.0f)
                         : 1.0f / (gdist + 1e-6f);

    // ---- softmax over 11 agents ----
    float cmax = combined;
#pragma unroll
    for (int off = 8; off >= 1; off >>= 1)
        cmax = fmaxf(cmax, __shfl_xor(cmax, off, LANES_PER_ROW));

    const float e = (lane < N_AGENTS) ? expf(combined - cmax) : 0.0f;

#if USE_ASYNC_LDS
    __builtin_amdgcn_s_wait_asynccnt(0);
    const float qa = (lane < N_AGENTS) ? q_s[grp][lane] : 0.0f;
#else
    const float qa = (lane < N_AGENTS)
                   ? q[(size_t)row * N_AGENTS + lane] : 0.0f;
#endif

    float esum = e;
    float tot  = qa * e;
#pragma unroll
    for (int off = 8; off >= 1; off >>= 1) {
        esum += __shfl_xor(esum, off, LANES_PER_ROW);
        tot  += __shfl_xor(tot,  off, LANES_PER_ROW);
    }

    if (lane == 0)
        out[row] = (float)N_AGENTS * tot / esum;
}

extern "C" void kernel_launch(void* const* d_in, const int* in_sizes, int n_in,
                              void* d_out, int out_size, void* d_ws, size_t ws_size,
                              hipStream_t stream) {
    (void)n_in; (void)out_size; (void)d_ws; (void)ws_size;
    const float* q  = (const float*)d_in[0];   // agents_q  (a,b,11) f32
    const float* gs = (const float*)d_in[1];   // global_state (a,b,1265) f32
    float* out = (float*)d_out;                // (a,b,1) f32

    const int B = in_sizes[0] / N_AGENTS;      // 65536 rows
    const int blocks = (B + ROWS_PER_BLOCK - 1) / ROWS_PER_BLOCK;
    qmix_fused_kernel<<<blocks, THREADS, 0, stream>>>(q, gs, out, B);
}